// KernelProduct_56392920597050
// MI455X (gfx1250) — compile-verified
//
#include <hip/hip_runtime.h>

// Problem constants (from reference): B=2048, F=40, E=64, I = F*(F-1)/2 = 780
#define BK 2048
#define FK 40
#define EK 64
#define IK 780
#define NTILES ((IK + 15) / 16)   // 49 pair-tiles of 16

typedef __attribute__((ext_vector_type(2))) float v2f;
typedef __attribute__((ext_vector_type(8))) float v8f;

// ---------------------------------------------------------------------------
// Kernel 1: ksum[p*EK+e] = sum_k kernel[k*IK*EK + p*EK + e]   (reads 12.8 MB once)
//           plus pair index tables ii[p], jj[p].
// Consecutive threads touch consecutive addresses -> fully coalesced.
// ---------------------------------------------------------------------------
__global__ __launch_bounds__(256) void prep_kernel(const float* __restrict__ kern,
                                                   float* __restrict__ ksum,
                                                   int* __restrict__ ii,
                                                   int* __restrict__ jj) {
    int t = blockIdx.x * blockDim.x + threadIdx.x;
    if (t < IK * EK) {
        float s = 0.0f;
        #pragma unroll 4
        for (int k = 0; k < EK; ++k) s += kern[k * (IK * EK) + t];
        ksum[t] = s;
    }
    if (t < IK) {
        int rem = t, i = 0;
        while (rem >= FK - 1 - i) { rem -= FK - 1 - i; ++i; }
        ii[t] = i;
        jj[t] = i + 1 + rem;
    }
}

// ---------------------------------------------------------------------------
// Kernel 2: one block per batch. Stage inputs[b] (40x64 f32, 10 KB) in LDS once;
// 8 waves sweep 16-pair tiles. Per tile, 16 accumulating V_WMMA_F32_16X16X4_F32
// compute D = (X_i * ksum) x (X_j)^T over K=64; out[b,p] = D[p,p].
//
// A 16x4 f32 layout (ISA 7.12.2): lanes 0-15 -> K={0,1} in v[0],v[1];
//                                 lanes 16-31 -> K={2,3}.  B 4x16 mirrored.
// C/D 16x16 f32 layout: VGPR r = row r (lanes 0-15) / row r+8 (lanes 16-31),
//                       col = lane & 15.  Diagonal p<8 -> lane p, c[p];
//                       p>=8 -> lane p+16, c[p-8].
// ---------------------------------------------------------------------------
__global__ __launch_bounds__(256) void interact_kernel(const float* __restrict__ in,
                                                       const float* __restrict__ ksum,
                                                       const int* __restrict__ ii,
                                                       const int* __restrict__ jj,
                                                       float* __restrict__ out) {
    __shared__ float s_in[FK * EK];  // 2560 floats = 10 KB (well under 320 KB/WGP)

    const int b = blockIdx.x;
    const float* inb = in + (size_t)b * (FK * EK);
    for (int t = threadIdx.x; t < FK * EK; t += 256) s_in[t] = inb[t];
    __syncthreads();

    const int lane = threadIdx.x & 31;
    const int wave = threadIdx.x >> 5;
    const int ka0  = (lane < 16) ? 0 : 2;  // K offset for this half-wave (A & B)
    const int pr   = lane & 15;            // pair-in-tile for both A row and B col

    for (int t = wave; t < NTILES; t += 8) {          // wave-uniform: EXEC stays full
        int p  = t * 16 + pr;
        int pc = (p < IK) ? p : (IK - 1);             // clamp pads; never stored
        const int fi = ii[pc];
        const int fj = jj[pc];
        const float* __restrict__ ks = ksum + (size_t)pc * EK;
        const float* __restrict__ si = s_in + fi * EK;
        const float* __restrict__ sj = s_in + fj * EK;

        v8f c = {};
        #pragma unroll
        for (int eb = 0; eb < EK; eb += 4) {
            v2f a, bm;
            a.x  = si[eb + ka0]     * ks[eb + ka0];
            a.y  = si[eb + ka0 + 1] * ks[eb + ka0 + 1];
            bm.x = sj[eb + ka0];
            bm.y = sj[eb + ka0 + 1];
            c = __builtin_amdgcn_wmma_f32_16x16x4_f32(
                    /*neg_a=*/false, a, /*neg_b=*/false, bm,
                    /*c_mod=*/(short)0, c, /*reuse_a=*/false, /*reuse_b=*/false);
        }

        // Diagonal extraction + store (masked; after all WMMAs of this tile).
        int  sel  = 0, pout = 0;
        bool has  = false;
        if (lane < 8)                   { sel = lane;      pout = t * 16 + lane;          has = true; }
        else if (lane >= 16 && lane < 24) { sel = lane - 16; pout = t * 16 + (lane - 16) + 8; has = true; }
        if (has && pout < IK) out[(size_t)b * IK + pout] = c[sel];
    }
}

// ---------------------------------------------------------------------------
// Launcher. d_in[0]=inputs (B*F*E f32), d_in[1]=kernel (E*I*E f32),
// d_in[2]=training (unused). d_out = B*I f32. Workspace: ksum | ii | jj.
// ---------------------------------------------------------------------------
extern "C" void kernel_launch(void* const* d_in, const int* in_sizes, int n_in,
                              void* d_out, int out_size, void* d_ws, size_t ws_size,
                              hipStream_t stream) {
    (void)in_sizes; (void)n_in; (void)out_size; (void)ws_size;

    const float* inputs = (const float*)d_in[0];
    const float* kern   = (const float*)d_in[1];

    float* ksum = (float*)d_ws;                                   // IK*EK floats
    int*   ii   = (int*)((char*)d_ws + (size_t)IK * EK * sizeof(float));
    int*   jj   = ii + IK;

    const int n1 = IK * EK;  // 49920 >= IK, covers both prep tasks
    prep_kernel<<<(n1 + 255) / 256, 256, 0, stream>>>(kern, ksum, ii, jj);

    interact_kernel<<<BK, 256, 0, stream>>>(inputs, ksum, ii, jj, (float*)d_out);
}